// GNN_45767171506444
// MI455X (gfx1250) — compile-verified
//
#include <hip/hip_runtime.h>

typedef float v2f __attribute__((ext_vector_type(2)));
typedef float v8f __attribute__((ext_vector_type(8)));

#define N_NODES  50000
#define N_EDGES  600000
#define IN_C     128
#define HID      128
#define OUT_C    32
#define N_GRAPHS 64
#define NODE_PAD 50048      // N_NODES rounded up to multiple of 128
#define EDGE_PAD 600064     // N_EDGES rounded up
#define SCAN_BLOCKS 196     // ceil(50000/256)

// ---------------------------------------------------------------------------
// WMMA helper: D = A(16x4,f32) x B(4x16,f32) + C(16x16,f32)
// Fragment layout (CDNA5 ISA 7.12.2):
//   A: lane L holds A[L&15][k0+2*(L>>4) + j] in vgpr j   (j = 0,1)
//   B: lane L holds B[k0+2*(L>>4) + j][L&15] in vgpr j   (B = W^T -> W row-major)
//   C: vgpr v, lane L -> C[v + 8*(L>>4)][L&15]
// ---------------------------------------------------------------------------
__device__ __forceinline__ v8f wmma_f32_16x16x4(v2f a, v2f b, v8f c) {
  return __builtin_amdgcn_wmma_f32_16x16x4_f32(
      /*neg_a=*/false, a, /*neg_b=*/false, b,
      /*c_mod=*/(short)0, c, /*reuse_a=*/false, /*reuse_b=*/false);
}

// ---------------------------------------------------------------------------
// Zero an int buffer (grid-stride)
// ---------------------------------------------------------------------------
__global__ void zero_u32(unsigned int* __restrict__ p, int n) {
  int i = blockIdx.x * blockDim.x + threadIdx.x;
  int stride = gridDim.x * blockDim.x;
  for (; i < n; i += stride) p[i] = 0u;
}

// ---------------------------------------------------------------------------
// CSR build step 1: in-degree histogram
// ---------------------------------------------------------------------------
__global__ void __launch_bounds__(256)
count_deg_kernel(const int* __restrict__ dst, int* __restrict__ deg, int n_edges) {
  int e = blockIdx.x * blockDim.x + threadIdx.x;
  if (e >= n_edges) return;
  atomicAdd(&deg[dst[e]], 1);
}

// ---------------------------------------------------------------------------
// CSR build step 2a: per-block exclusive scan (Hillis-Steele in LDS),
// emits per-block totals.
// ---------------------------------------------------------------------------
__global__ void __launch_bounds__(256)
scan_block_kernel(const int* __restrict__ deg, int* __restrict__ row_start,
                  int* __restrict__ bsums, int n) {
  __shared__ int sh[256];
  const int tid = threadIdx.x;
  const int i = blockIdx.x * 256 + tid;
  const int v = (i < n) ? deg[i] : 0;
  sh[tid] = v;
  __syncthreads();
#pragma unroll
  for (int off = 1; off < 256; off <<= 1) {
    int t = (tid >= off) ? sh[tid - off] : 0;
    __syncthreads();
    sh[tid] += t;
    __syncthreads();
  }
  if (i < n) row_start[i] = sh[tid] - v;   // exclusive
  if (tid == 255) bsums[blockIdx.x] = sh[255];
}

// ---------------------------------------------------------------------------
// CSR build step 2b: exclusive scan of the block totals (single block)
// ---------------------------------------------------------------------------
__global__ void __launch_bounds__(256)
scan_sums_kernel(int* __restrict__ bsums, int nblocks) {
  __shared__ int sh[256];
  const int tid = threadIdx.x;
  const int v = (tid < nblocks) ? bsums[tid] : 0;
  sh[tid] = v;
  __syncthreads();
#pragma unroll
  for (int off = 1; off < 256; off <<= 1) {
    int t = (tid >= off) ? sh[tid - off] : 0;
    __syncthreads();
    sh[tid] += t;
    __syncthreads();
  }
  if (tid < nblocks) bsums[tid] = sh[tid] - v;  // exclusive
}

// ---------------------------------------------------------------------------
// CSR build step 2c: add block offsets; initialize fill cursors
// ---------------------------------------------------------------------------
__global__ void __launch_bounds__(256)
scan_add_kernel(int* __restrict__ row_start, int* __restrict__ cursor,
                const int* __restrict__ bsums, int n) {
  int i = blockIdx.x * 256 + threadIdx.x;
  if (i >= n) return;
  int r = row_start[i] + bsums[blockIdx.x];
  row_start[i] = r;
  cursor[i] = r;
}

// ---------------------------------------------------------------------------
// CSR build step 3: counting-sort src ids into CSR slots
// ---------------------------------------------------------------------------
__global__ void __launch_bounds__(256)
fill_csr_kernel(const int* __restrict__ src, const int* __restrict__ dst,
                int* __restrict__ cursor, int* __restrict__ csr_src, int n_edges) {
  int e = blockIdx.x * blockDim.x + threadIdx.x;
  if (e >= n_edges) return;
  int pos = atomicAdd(&cursor[dst[e]], 1);
  csr_src[pos] = src[e];
}

// ---------------------------------------------------------------------------
// Mean aggregation by gather: one wave per destination node, lane = float4
// chunk of the 128 channels. Zero floating-point atomics.
// ---------------------------------------------------------------------------
__global__ void __launch_bounds__(256)
aggregate_mean_kernel(const float* __restrict__ feat,
                      const int* __restrict__ row_start,
                      const int* __restrict__ csr_src,
                      float* __restrict__ mean, int n_nodes) {
  int idx = blockIdx.x * blockDim.x + threadIdx.x;
  int node = idx >> 5;
  if (node >= n_nodes) return;
  int ch = (idx & 31) << 2;
  int start = row_start[node];
  int end = (node + 1 < n_nodes) ? row_start[node + 1] : N_EDGES;

  float4 acc = make_float4(0.f, 0.f, 0.f, 0.f);
  for (int j = start; j < end; ++j) {
    int s = csr_src[j];
    const float4 v = *(const float4*)(feat + (size_t)s * HID + ch);
    acc.x += v.x; acc.y += v.y; acc.z += v.z; acc.w += v.w;
  }
  float invd = 1.0f / fmaxf((float)(end - start), 1.0f);
  acc.x *= invd; acc.y *= invd; acc.z *= invd; acc.w *= invd;
  *(float4*)(mean + (size_t)node * HID + ch) = acc;
}

// ---------------------------------------------------------------------------
// Fused SAGEConv GEMM: out = relu( mean @ Wl^T + bias + xin @ Wr^T )
// Grid: N_NODES/16 blocks of 256 threads (8 waves = 8 n-tiles).
// Each wave: one 16x16 tile via 64 V_WMMA_F32_16X16X4_F32 (pure load+wmma).
// ---------------------------------------------------------------------------
__global__ void __launch_bounds__(256)
sage_gemm_kernel(const float* __restrict__ mean, const float* __restrict__ xin,
                 const float* __restrict__ Wl, const float* __restrict__ bias,
                 const float* __restrict__ Wr,
                 float* __restrict__ out, int do_relu) {
  const int wave = threadIdx.x >> 5;   // 0..7 -> n-tile
  const int lane = threadIdx.x & 31;
  const int half = lane >> 4;
  const int lm   = lane & 15;
  const int mbase = blockIdx.x * 16;
  const int nbase = wave * 16;

  const int row_a = mbase + lm;        // node row feeding the A fragment
  const int row_b = nbase + lm;        // weight row feeding the B fragment

  v8f c = {};

  // ---- pass 1: mean-neighbor features @ Wl^T ----
  {
    const float* Arow = mean + (size_t)row_a * HID + 2 * half;
    const float* Brow = Wl   + (size_t)row_b * HID + 2 * half;
#pragma unroll
    for (int k0 = 0; k0 < HID; k0 += 4) {
      v2f a; a.x = Arow[k0]; a.y = Arow[k0 + 1];
      v2f b; b.x = Brow[k0]; b.y = Brow[k0 + 1];
      c = wmma_f32_16x16x4(a, b, c);
    }
  }
  // ---- pass 2: self features @ Wr^T ----
  {
    const float* Arow = xin + (size_t)row_a * HID + 2 * half;
    const float* Brow = Wr  + (size_t)row_b * HID + 2 * half;
#pragma unroll
    for (int k0 = 0; k0 < HID; k0 += 4) {
      v2f a; a.x = Arow[k0]; a.y = Arow[k0 + 1];
      v2f b; b.x = Brow[k0]; b.y = Brow[k0 + 1];
      c = wmma_f32_16x16x4(a, b, c);
    }
  }

  const float bv = bias[nbase + lm];
#pragma unroll
  for (int v = 0; v < 8; ++v) {
    float o = c[v] + bv;
    if (do_relu) o = fmaxf(o, 0.0f);
    const int mrow = mbase + v + 8 * half;
    out[(size_t)mrow * HID + nbase + lm] = o;
  }
}

// ---------------------------------------------------------------------------
// Global mean pool: batch is SORTED, so each graph's rows are contiguous.
// 64 blocks x 128 threads; block g binary-searches its row range and
// accumulates column-per-thread. Fully coalesced, no atomics.
// ---------------------------------------------------------------------------
__device__ __forceinline__ int lower_bound_i(const int* __restrict__ a, int n, int key) {
  int lo = 0, hi = n;
  while (lo < hi) {
    int mid = (lo + hi) >> 1;
    if (a[mid] < key) lo = mid + 1; else hi = mid;
  }
  return lo;
}

__global__ void __launch_bounds__(128)
pool_kernel(const float* __restrict__ h, const int* __restrict__ batch,
            float* __restrict__ pooled) {
  const int g = blockIdx.x;
  const int t = threadIdx.x;      // channel
  const int lo = lower_bound_i(batch, N_NODES, g);
  const int hi = lower_bound_i(batch, N_NODES, g + 1);
  float acc = 0.0f;
  for (int r = lo; r < hi; ++r) acc += h[(size_t)r * HID + t];
  const float inv = 1.0f / fmaxf((float)(hi - lo), 1.0f);
  pooled[(size_t)g * HID + t] = acc * inv;
}

// ---------------------------------------------------------------------------
// Final classifier: out[64,32] = pooled @ Wlin^T + blin
// One block of 256 threads: 8 waves cover the 4x2 grid of 16x16 tiles.
// ---------------------------------------------------------------------------
__global__ void __launch_bounds__(256)
final_gemm_kernel(const float* __restrict__ pooled,
                  const float* __restrict__ Wlin, const float* __restrict__ blin,
                  float* __restrict__ out) {
  const int wave  = threadIdx.x >> 5;  // 0..7
  const int mtile = wave >> 1;         // 0..3  (graphs)
  const int ntile = wave & 1;          // 0..1  (output channels)
  const int lane = threadIdx.x & 31;
  const int half = lane >> 4;
  const int lm   = lane & 15;

  const int row_a = mtile * 16 + lm;
  const int row_b = ntile * 16 + lm;

  v8f c = {};
  const float* Arow = pooled + (size_t)row_a * HID + 2 * half;
  const float* Brow = Wlin   + (size_t)row_b * HID + 2 * half;
#pragma unroll
  for (int k0 = 0; k0 < HID; k0 += 4) {
    v2f a; a.x = Arow[k0]; a.y = Arow[k0 + 1];
    v2f b; b.x = Brow[k0]; b.y = Brow[k0 + 1];
    c = wmma_f32_16x16x4(a, b, c);
  }

  const float bv = blin[ntile * 16 + lm];
#pragma unroll
  for (int v = 0; v < 8; ++v) {
    const int g = mtile * 16 + v + 8 * half;
    out[(size_t)g * OUT_C + ntile * 16 + lm] = c[v] + bv;
  }
}

// ---------------------------------------------------------------------------
extern "C" void kernel_launch(void* const* d_in, const int* in_sizes, int n_in,
                              void* d_out, int out_size, void* d_ws, size_t ws_size,
                              hipStream_t stream) {
  (void)in_sizes; (void)n_in; (void)out_size; (void)ws_size;

  const float* x     = (const float*)d_in[0];
  const int*   ei    = (const int*)d_in[1];
  const int*   batch = (const int*)d_in[2];
  const float* W1l   = (const float*)d_in[3];
  const float* b1    = (const float*)d_in[4];
  const float* W1r   = (const float*)d_in[5];
  const float* W2l   = (const float*)d_in[6];
  const float* b2    = (const float*)d_in[7];
  const float* W2r   = (const float*)d_in[8];
  const float* Wlin  = (const float*)d_in[9];
  const float* blin  = (const float*)d_in[10];
  float* out = (float*)d_out;

  // Workspace layout: ints (CSR) then floats. ~80 MB total.
  int*   deg_i   = (int*)d_ws;                 // NODE_PAD
  int*   cursor  = deg_i   + NODE_PAD;         // NODE_PAD
  int*   row_s   = cursor  + NODE_PAD;         // NODE_PAD
  int*   bsums   = row_s   + NODE_PAD;         // 256
  int*   csr_src = bsums   + 256;              // EDGE_PAD
  float* mean    = (float*)(csr_src + EDGE_PAD);
  float* h1      = mean + (size_t)N_NODES * HID;
  float* h2      = h1   + (size_t)N_NODES * HID;
  float* pooled  = h2   + (size_t)N_NODES * HID;   // N_GRAPHS*HID

  const int* src = ei;
  const int* dst = ei + N_EDGES;

  const dim3 blk(256);
  const int edge_blocks = (N_EDGES + 255) / 256;       // 2344
  const int agg_blocks  = (N_NODES * 32 + 255) / 256;  // 6250
  const int gemm_blocks = N_NODES / 16;                // 3125

  // ---- build CSR (shared by both layers) ----
  zero_u32<<<64, blk, 0, stream>>>((unsigned int*)deg_i, NODE_PAD);
  count_deg_kernel<<<edge_blocks, blk, 0, stream>>>(dst, deg_i, N_EDGES);
  scan_block_kernel<<<SCAN_BLOCKS, blk, 0, stream>>>(deg_i, row_s, bsums, N_NODES);
  scan_sums_kernel<<<1, blk, 0, stream>>>(bsums, SCAN_BLOCKS);
  scan_add_kernel<<<SCAN_BLOCKS, blk, 0, stream>>>(row_s, cursor, bsums, N_NODES);
  fill_csr_kernel<<<edge_blocks, blk, 0, stream>>>(src, dst, cursor, csr_src, N_EDGES);

  // ---- layer 1 ----
  aggregate_mean_kernel<<<agg_blocks, blk, 0, stream>>>(x, row_s, csr_src, mean, N_NODES);
  sage_gemm_kernel<<<gemm_blocks, blk, 0, stream>>>(mean, x, W1l, b1, W1r, h1, 1);

  // ---- layer 2 ----
  aggregate_mean_kernel<<<agg_blocks, blk, 0, stream>>>(h1, row_s, csr_src, mean, N_NODES);
  sage_gemm_kernel<<<gemm_blocks, blk, 0, stream>>>(mean, h1, W2l, b2, W2r, h2, 1);

  // ---- global mean pool + classifier ----
  pool_kernel<<<N_GRAPHS, dim3(128), 0, stream>>>(h2, batch, pooled);
  final_gemm_kernel<<<1, blk, 0, stream>>>(pooled, Wlin, blin, out);
}